// Sampler_39883066311048
// MI455X (gfx1250) — compile-verified
//
#include <hip/hip_runtime.h>
#include <hip/hip_bf16.h>
#include <cfloat>
#include <cstdint>

// MI455X (gfx1250) sampler:
//  - wave32, 512 threads/block = 16 waves, one block per batch row
//  - Phase 1: HBM streaming pass via async global->LDS double buffering
//  - Selection: value-space bisection, all passes L2-resident (131MB < 192MB L2)
//  - f32 block reductions ride V_WMMA_F32_16X16X4_F32 (ones-matrix rowsum trick)

#define TPB    512
#define NWAVES (TPB / 32)
#define CAND   64

typedef __attribute__((ext_vector_type(2))) float v2f;
typedef __attribute__((ext_vector_type(8))) float v8f;

#if defined(__HIP_DEVICE_COMPILE__)
#if __has_builtin(__builtin_amdgcn_wmma_f32_16x16x4_f32)
#define USE_WMMA_SUM 1
#endif
#if __has_builtin(__builtin_amdgcn_global_load_async_to_lds_b128) && \
    __has_builtin(__builtin_amdgcn_s_wait_asynccnt)
#define USE_ASYNC_LDS 1
// Builtin parameter types: AS(1)/AS(3) pointers to <4 x i32> (vector_size 16).
typedef int v4i_vs __attribute__((vector_size(16)));
typedef __attribute__((address_space(1))) v4i_vs* as1_v4i;
typedef __attribute__((address_space(3))) v4i_vs* as3_v4i;
#endif
#endif

// ---------------- wave / block reductions ----------------

__device__ __forceinline__ float waveSumF(float x) {
#if defined(USE_WMMA_SUM)
  // D[M][N] = sum_k A[M][k] * 1 ; A vgpr0 = x (K=0 lanes0-15, K=2 lanes16-31),
  // vgpr1 = 0. Lane L<16: sum_j c[j] = sum(x0..x7 + x16..x23); L>=16: complement.
  v2f a; a[0] = x;    a[1] = 0.0f;
  v2f b; b[0] = 1.0f; b[1] = 1.0f;
  v8f c = {};
  c = __builtin_amdgcn_wmma_f32_16x16x4_f32(false, a, false, b, (short)0, c,
                                            false, false);
  float s = 0.0f;
#pragma unroll
  for (int j = 0; j < 8; ++j) s += c[j];
  s += __shfl_xor(s, 16, 32);
  return s;
#else
#pragma unroll
  for (int off = 16; off; off >>= 1) x += __shfl_xor(x, off, 32);
  return x;
#endif
}

__device__ __forceinline__ int waveSumI(int x) {
#pragma unroll
  for (int off = 16; off; off >>= 1) x += __shfl_xor(x, off, 32);
  return x;
}

__device__ __forceinline__ float blockSumF(float x, float* sred) {
  const int lane = threadIdx.x & 31;
  const int wid  = threadIdx.x >> 5;
  float s = waveSumF(x);
  __syncthreads();
  if (lane == 0) sred[wid] = s;
  __syncthreads();
  if (wid == 0) {
    float v = (lane < NWAVES) ? sred[lane] : 0.0f;
    v = waveSumF(v);
    if (lane == 0) sred[0] = v;
  }
  __syncthreads();
  return sred[0];
}

__device__ __forceinline__ int blockSumI(int x, int* sred) {
  const int lane = threadIdx.x & 31;
  const int wid  = threadIdx.x >> 5;
  int s = waveSumI(x);
  __syncthreads();
  if (lane == 0) sred[wid] = s;
  __syncthreads();
  if (wid == 0) {
    int v = (lane < NWAVES) ? sred[lane] : 0;
    v = waveSumI(v);
    if (lane == 0) sred[0] = v;
  }
  __syncthreads();
  return sred[0];
}

// argmax with first-occurrence (lowest index) tie-break, broadcast to all
__device__ __forceinline__ void blockArgMax(float v, int i, float* sv, int* si,
                                            float& outV, int& outI) {
  const int lane = threadIdx.x & 31;
  const int wid  = threadIdx.x >> 5;
#pragma unroll
  for (int off = 16; off; off >>= 1) {
    float ov = __shfl_xor(v, off, 32);
    int   oi = __shfl_xor(i, off, 32);
    if (ov > v || (ov == v && oi < i)) { v = ov; i = oi; }
  }
  __syncthreads();
  if (lane == 0) { sv[wid] = v; si[wid] = i; }
  __syncthreads();
  if (wid == 0) {
    float vv = (lane < NWAVES) ? sv[lane] : -FLT_MAX;
    int   ii = (lane < NWAVES) ? si[lane] : 0x7fffffff;
#pragma unroll
    for (int off = 16; off; off >>= 1) {
      float ov = __shfl_xor(vv, off, 32);
      int   oi = __shfl_xor(ii, off, 32);
      if (ov > vv || (ov == vv && oi < ii)) { vv = ov; ii = oi; }
    }
    if (lane == 0) { sv[0] = vv; si[0] = ii; }
  }
  __syncthreads();
  outV = sv[0];
  outI = si[0];
}

// ---------------- sampler kernel ----------------

__global__ __launch_bounds__(TPB, 1) void sampler_kernel(
    const float* __restrict__ logits, const float* __restrict__ temperature,
    const float* __restrict__ top_p, const float* __restrict__ noise,
    const long long* __restrict__ top_k, int* __restrict__ out_sampled,
    int* __restrict__ out_indices, float* __restrict__ out_lp,
    int* __restrict__ out_ranks, int V, int K) {
  __shared__ float s_redf[NWAVES];
  __shared__ int   s_redi[NWAVES];
  __shared__ float s_av[NWAVES];
  __shared__ int   s_ai[NWAVES];
  __shared__ int   s_cnt;
  __shared__ float s_cv[CAND];
  __shared__ int   s_ci[CAND];
#if defined(USE_ASYNC_LDS)
  __shared__ __align__(16) float s_stage[2][TPB * 4];
#endif

  const int row = blockIdx.x;
  const int tid = threadIdx.x;
  const float* __restrict__ rowL = logits + (size_t)row * (size_t)V;
  const float* __restrict__ rowU = noise + (size_t)row * (size_t)V;
  const bool vec4 = ((V & 3) == 0);

  // ---- Phase 1: streaming pass (HBM) — max/argmax, min, online sum-exp ----
  float thrMax = -FLT_MAX, thrSum = 0.0f, thrMin = FLT_MAX;
  int   thrArg = 0;

#if defined(USE_ASYNC_LDS)
  if (vec4) {
    const int TILE   = TPB * 4;
    const int ntiles = (V + TILE - 1) / TILE;
    {  // prime tile 0
      const int base = tid * 4;
      if (base < V)
        __builtin_amdgcn_global_load_async_to_lds_b128(
            (as1_v4i)(const void*)(rowL + base),
            (as3_v4i)(void*)&s_stage[0][tid * 4], 0, 0);
    }
    for (int t = 0; t < ntiles; ++t) {
      const int cur = t & 1;
      if (t + 1 < ntiles) {
        const int nb = (t + 1) * TILE + tid * 4;
        if (nb < V)
          __builtin_amdgcn_global_load_async_to_lds_b128(
              (as1_v4i)(const void*)(rowL + nb),
              (as3_v4i)(void*)&s_stage[cur ^ 1][tid * 4], 0, 0);
        __builtin_amdgcn_s_wait_asynccnt(1);
      } else {
        __builtin_amdgcn_s_wait_asynccnt(0);
      }
      __syncthreads();
      const int base = t * TILE + tid * 4;
      if (base < V) {
#pragma unroll
        for (int j = 0; j < 4; ++j) {
          const float x = s_stage[cur][tid * 4 + j];
          thrMin = fminf(thrMin, x);
          if (x > thrMax) {
            thrSum = thrSum * __expf(thrMax - x) + 1.0f;
            thrMax = x;
            thrArg = base + j;
          } else {
            thrSum += __expf(x - thrMax);
          }
        }
      }
      __syncthreads();
    }
  } else
#endif
  {
    for (int i = tid; i < V; i += TPB) {
      const float x = rowL[i];
      thrMin = fminf(thrMin, x);
      if (x > thrMax) {
        thrSum = thrSum * __expf(thrMax - x) + 1.0f;
        thrMax = x;
        thrArg = i;
      } else {
        thrSum += __expf(x - thrMax);
      }
    }
  }

  float M;
  int   greedyIdx;
  blockArgMax(thrMax, thrArg, s_av, s_ai, M, greedyIdx);
  const float sumExpRaw = blockSumF(thrSum * __expf(thrMax - M), s_redf);
  float mnNeg;
  int   dummyI;
  blockArgMax(-thrMin, 0, s_av, s_ai, mnNeg, dummyI);
  const float mn = -mnNeg;

  // ---- Per-row scalars ----
  const float     T     = temperature[row];
  const float     pp    = top_p[row];
  const long long kkll  = top_k[row];
  const bool      greedy = (T < 1e-5f);
  const float     Tsafe  = greedy ? 1.0f : T;
  const float     invT   = 1.0f / Tsafe;
  const int k_eff = (kkll <= 0 || kkll >= (long long)V) ? V : (int)kkll;

  // ---- Phase 2: fused bisection for k-th largest (t_k) and K-th (t20) ----
  const float hi0 = M + fmaxf(fabsf(M) * 1e-6f, 1e-6f);
  float lo1 = mn - 1.0f, hi1 = hi0;  // top-k threshold (x-space; T>0 monotone)
  float lo2 = mn - 1.0f, hi2 = hi0;  // top-K logprob threshold
  for (int it = 0; it < 32; ++it) {
    const float mid1 = 0.5f * (lo1 + hi1);
    const float mid2 = 0.5f * (lo2 + hi2);
    int c1 = 0, c2 = 0;
    if (vec4) {
      const float4* r4 = (const float4*)rowL;
      const int V4 = V >> 2;
      for (int i = tid; i < V4; i += TPB) {
        const float4 x = r4[i];
        c1 += (x.x >= mid1) + (x.y >= mid1) + (x.z >= mid1) + (x.w >= mid1);
        c2 += (x.x >= mid2) + (x.y >= mid2) + (x.z >= mid2) + (x.w >= mid2);
      }
    } else {
      for (int i = tid; i < V; i += TPB) {
        const float x = rowL[i];
        c1 += (x >= mid1);
        c2 += (x >= mid2);
      }
    }
    c1 = blockSumI(c1, s_redi);
    c2 = blockSumI(c2, s_redi);
    if (c1 >= k_eff) lo1 = mid1; else hi1 = mid1;
    if (c2 >= K)     lo2 = mid2; else hi2 = mid2;
  }
  const float t_k = lo1;  // count(x >= t_k) == k_eff
  const float t20 = lo2;  // count(x >= t20) >= K (tight)

  // ---- Phase 3: top-p threshold + Gumbel-max sample (skipped if greedy) ----
  int sampled;
  if (greedy) {
    sampled = greedyIdx;
  } else {
    // Z_k = sum over kept-k of exp((x-M)/T)
    float zloc = 0.0f;
    for (int i = tid; i < V; i += TPB) {
      const float x = rowL[i];
      if (x >= t_k) zloc += __expf((x - M) * invT);
    }
    const float Zk     = blockSumF(zloc, s_redf);
    const float target = pp * Zk;
    // bisect smallest kept value: token x kept iff mass{x' > x} < p*Z_k
    float lo = mn - 1.0f, hi = hi0;
    for (int it = 0; it < 30; ++it) {
      const float mid  = 0.5f * (lo + hi);
      float gloc = 0.0f;
      if (vec4) {
        const float4* r4 = (const float4*)rowL;
        const int V4 = V >> 2;
        for (int i = tid; i < V4; i += TPB) {
          const float4 x = r4[i];
          if (x.x >= t_k && x.x > mid) gloc += __expf((x.x - M) * invT);
          if (x.y >= t_k && x.y > mid) gloc += __expf((x.y - M) * invT);
          if (x.z >= t_k && x.z > mid) gloc += __expf((x.z - M) * invT);
          if (x.w >= t_k && x.w > mid) gloc += __expf((x.w - M) * invT);
        }
      } else {
        for (int i = tid; i < V; i += TPB) {
          const float x = rowL[i];
          if (x >= t_k && x > mid) gloc += __expf((x - M) * invT);
        }
      }
      const float g = blockSumF(gloc, s_redf);
      if (g < target) hi = mid; else lo = mid;
    }
    const float t_f = fmaxf(t_k, hi);  // combined top-k/top-p threshold

    // Gumbel-max over filtered tokens: argmax(x/T - log(-log(u)))
    float bv = -FLT_MAX;
    int   bi = 0x7fffffff;
    for (int i = tid; i < V; i += TPB) {
      const float x = rowL[i];
      if (x >= t_f) {
        const float u   = rowU[i];
        const float val = x * invT - __logf(-__logf(u));
        if (val > bv || (val == bv && i < bi)) { bv = val; bi = i; }
      }
    }
    float dumv;
    blockArgMax(bv, bi, s_av, s_ai, dumv, sampled);
  }

  // ---- Phase 4: fused rank count + top-K candidate collection ----
  const float xs = rowL[sampled];  // uniform broadcast load
  if (tid == 0) s_cnt = 0;
  __syncthreads();
  int crank = 0;
  for (int i = tid; i < V; i += TPB) {
    const float x = rowL[i];
    crank += (x >= xs);
    if (x >= t20) {
      const int p = atomicAdd(&s_cnt, 1);
      if (p < CAND) { s_cv[p] = x; s_ci[p] = i; }
    }
  }
  const int rank = blockSumI(crank, s_redi);  // barrier => s_cnt/s_cv visible

  // ---- Phase 5: tiny sort + write-out ----
  if (tid == 0) {
    int n = s_cnt < CAND ? s_cnt : CAND;
    if (n < 1) { n = 1; s_cv[0] = M; s_ci[0] = greedyIdx; }
    for (int a = 1; a < n; ++a) {  // insertion sort: value desc, index asc
      const float v  = s_cv[a];
      const int   ii = s_ci[a];
      int b = a - 1;
      while (b >= 0 && (s_cv[b] < v || (s_cv[b] == v && s_ci[b] > ii))) {
        s_cv[b + 1] = s_cv[b];
        s_ci[b + 1] = s_ci[b];
        --b;
      }
      s_cv[b + 1] = v;
      s_ci[b + 1] = ii;
    }
    const float logZ     = __logf(sumExpRaw);
    const float token_lp = xs - M - logZ;
    out_sampled[row] = sampled;
    out_ranks[row]   = rank;
    int*   idxRow = out_indices + (size_t)row * (K + 1);
    float* lpRow  = out_lp + (size_t)row * (K + 1);
    idxRow[0] = sampled;
    lpRow[0]  = token_lp;
    for (int j = 0; j < K; ++j) {
      const int jj = (j < n) ? j : (n - 1);
      idxRow[1 + j] = s_ci[jj];
      lpRow[1 + j]  = s_cv[jj] - M - logZ;
    }
  }
}

// ---------------- host launch ----------------

extern "C" void kernel_launch(void* const* d_in, const int* in_sizes, int n_in,
                              void* d_out, int out_size, void* d_ws,
                              size_t ws_size, hipStream_t stream) {
  (void)n_in; (void)d_ws; (void)ws_size;
  const float*     logits      = (const float*)d_in[0];
  const float*     temperature = (const float*)d_in[1];
  const float*     top_p       = (const float*)d_in[2];
  const float*     noise       = (const float*)d_in[3];
  const long long* top_k       = (const long long*)d_in[4];  // jnp.int64

  const int B = in_sizes[1];          // temperature is [B]
  if (B <= 0) return;
  const int V = in_sizes[0] / B;      // logits is [B, V]
  // out = [B] i32 ++ [B,K+1] i32 ++ [B,K+1] f32 ++ [B] i32  => out/B = 2K+4
  int K = (out_size / B - 4) / 2;
  if (K < 1) K = 1;
  if (K > CAND - 1) K = CAND - 1;

  int*   out_sampled = (int*)d_out;
  int*   out_indices = out_sampled + B;
  float* out_lp      = (float*)(out_indices + (size_t)B * (K + 1));
  int*   out_ranks   = (int*)(out_lp + (size_t)B * (K + 1));

  sampler_kernel<<<dim3(B), dim3(TPB), 0, stream>>>(
      logits, temperature, top_p, noise, top_k, out_sampled, out_indices,
      out_lp, out_ranks, V, K);
}